// KalmanVAE_62886911148300
// MI455X (gfx1250) — compile-verified
//
#include <hip/hip_runtime.h>

// ---------------------------------------------------------------------------
// Mixture Kalman smoother (KVAE) for MI455X / gfx1250.
// One workgroup per batch chain (B=128). All per-step matrices live in LDS.
// GEMMs use V_WMMA_F32_16X16X4_F32 (f32 precision needed for the 80-step
// covariance recursion); inverses via workgroup Gauss-Jordan.
// ---------------------------------------------------------------------------

#define Bq 128
#define Tq 80
#define Aq 32
#define Zq 64
#define Kq 16
#define NTHREADS 256

typedef __attribute__((ext_vector_type(2))) float v2f;
typedef __attribute__((ext_vector_type(8))) float v8f;

#if defined(__AMDGCN__) && __has_builtin(__builtin_amdgcn_wmma_f32_16x16x4_f32)
#define USE_WMMA_F32 1
#else
#define USE_WMMA_F32 0
#endif

static __device__ __forceinline__ float ldA_el(const float* P, int ld, int m, int k, bool tr) {
  return tr ? P[k * ld + m] : P[m * ld + k];
}
static __device__ __forceinline__ float ldB_el(const float* P, int ld, int k, int n, bool tr) {
  return tr ? P[n * ld + k] : P[k * ld + n];
}

// D[MxN] = A[MxK] * B[KxN] from/to LDS. M,N multiples of 16, KD multiple of 4.
// Tiles distributed round-robin across the 8 waves; condition is wave-uniform
// so EXEC stays all-ones for every v_wmma issue (ISA 7.12 requirement).
static __device__ void mm(int M, int N, int KD,
                          const float* A, int lda, bool tA,
                          const float* B, int ldb, bool tB,
                          float* D, int ldd) {
  __syncthreads();
  const int lane = threadIdx.x & 31;
  const int wave = threadIdx.x >> 5;
  const int nw   = NTHREADS >> 5;
  const int tn = N >> 4;
  const int ntiles = (M >> 4) * tn;
  const int m16  = lane & 15;
  const int half = lane >> 4;  // lane group: rows r / r+8 of the C/D fragment
  for (int tile = wave; tile < ntiles; tile += nw) {
    const int bi = (tile / tn) << 4;
    const int bj = (tile % tn) << 4;
    v8f acc = {0.f, 0.f, 0.f, 0.f, 0.f, 0.f, 0.f, 0.f};
#if USE_WMMA_F32
    for (int kb = 0; kb < KD; kb += 4) {
      const int k0 = kb + half * 2;  // ISA: lanes 0-15 hold K=0,1; 16-31 hold K=2,3
      v2f af, bf;
      af.x = ldA_el(A, lda, bi + m16, k0,     tA);
      af.y = ldA_el(A, lda, bi + m16, k0 + 1, tA);
      bf.x = ldB_el(B, ldb, k0,     bj + m16, tB);
      bf.y = ldB_el(B, ldb, k0 + 1, bj + m16, tB);
      acc = __builtin_amdgcn_wmma_f32_16x16x4_f32(
          false, af, false, bf, (short)0, acc, false, false);
    }
#else
    for (int kb = 0; kb < KD; ++kb) {
      const float bv = ldB_el(B, ldb, kb, bj + m16, tB);
#pragma unroll
      for (int r = 0; r < 8; ++r)
        acc[r] += ldA_el(A, lda, bi + r + 8 * half, kb, tA) * bv;
    }
#endif
    // D fragment layout: VGPR r -> row (r + 8*half), col m16
#pragma unroll
    for (int r = 0; r < 8; ++r)
      D[(bi + r + 8 * half) * ldd + (bj + m16)] = acc[r];
  }
  __syncthreads();
}

// In-place Gauss-Jordan inverse of n x n LDS matrix (SPD -> no pivoting).
static __device__ void gj_inverse(float* M, int n) {
  const int tid = threadIdx.x;
  for (int k = 0; k < n; ++k) {
    __syncthreads();
    const float p = 1.0f / M[k * n + k];
    // scale row k (except pivot column)
    for (int j = tid; j < n; j += NTHREADS)
      if (j != k) M[k * n + j] *= p;
    __syncthreads();
    // rank-1 elimination on all (i != k, j != k)
    for (int idx = tid; idx < n * n; idx += NTHREADS) {
      const int i = idx / n, j = idx % n;
      if (i != k && j != k) M[i * n + j] -= M[i * n + k] * M[k * n + j];
    }
    __syncthreads();
    // fix column k
    for (int i = tid; i < n; i += NTHREADS)
      if (i != k) M[i * n + k] = -M[i * n + k] * p;
    if (tid == 0) M[k * n + k] = p;
  }
  __syncthreads();
}

// y = M(n_out x n_in) * x
static __device__ void matvec(int n_out, int n_in, const float* M, int ld,
                              const float* x, float* y) {
  __syncthreads();
  for (int i = threadIdx.x; i < n_out; i += NTHREADS) {
    float s = 0.f;
    for (int j = 0; j < n_in; ++j) s += M[i * ld + j] * x[j];
    y[i] = s;
  }
  __syncthreads();
}

// dst[i] = sum_k w[k] * bank[k*stride + i]  (bank is L2-resident: 256KB total)
static __device__ void mix16(float* dst, const float* __restrict__ bank,
                             const float* w, int n) {
  __syncthreads();
  for (int i = threadIdx.x; i < n; i += NTHREADS) {
    float s = 0.f;
#pragma unroll
    for (int k = 0; k < Kq; ++k) s += w[k] * bank[k * n + i];
    dst[i] = s;
  }
  __syncthreads();
}

static __device__ void softmax16(const float* __restrict__ logits, float* w) {
  __syncthreads();
  if (threadIdx.x == 0) {
    float mx = logits[0];
    for (int k = 1; k < Kq; ++k) mx = fmaxf(mx, logits[k]);
    float e[Kq], s = 0.f;
    for (int k = 0; k < Kq; ++k) { e[k] = __expf(logits[k] - mx); s += e[k]; }
    const float inv = 1.f / s;
    for (int k = 0; k < Kq; ++k) w[k] = e[k] * inv;
  }
  __syncthreads();
}

// LDS partition (floats):
//  Acur 4096 | Anx 4096 | Cb 2048 | sigp 4096 | sigf 4096 | tmp 4096
//  Kg 2048   | sigs 4096 | vectors 400                     -> 29072 floats
#define SMEM_FLOATS (6 * 4096 + 2 * 2048 + 16 + 6 * 64)

__global__ void __launch_bounds__(NTHREADS)
kvae_kernel(const float* __restrict__ obs, const float* __restrict__ alpha,
            const float* __restrict__ A_bank, const float* __restrict__ C_bank,
            float* __restrict__ out) {
  extern __shared__ float sm[];
  float* Acur = sm;                  // 64x64
  float* Anx  = Acur + Zq * Zq;      // 64x64
  float* Cb   = Anx + Zq * Zq;       // 32x64
  float* sigp = Cb + Aq * Zq;        // 64x64
  float* sigf = sigp + Zq * Zq;      // 64x64
  float* tmp  = sigf + Zq * Zq;      // 64x64 scratch (tail holds 32x32 S)
  float* Kg   = tmp + Zq * Zq;       // 64x32
  float* sigs = Kg + Zq * Aq;        // 64x64 smoother carry
  float* vw      = sigs + Zq * Zq;   // 16
  float* vmuprev = vw + Kq;          // 64
  float* vAmu    = vmuprev + Zq;     // 64
  float* verr    = vAmu + Zq;        // 64
  float* vt      = verr + Zq;        // 64
  float* vmf     = vt + Zq;          // 64
  float* vmus    = vmf + Zq;         // 64

  const int b = blockIdx.x;
  const int tid = threadIdx.x;
  const float* alpha_b = alpha + (size_t)b * Tq * Kq;
  const float* obs_b   = obs + (size_t)b * Tq * Aq;
  float* mu_out  = out + (size_t)b * Tq * Zq;
  float* sig_out = out + (size_t)Bq * Tq * Zq + (size_t)b * Tq * Zq * Zq;

  // carry init: mu0 = 0, sig0 = 20*I
  for (int i = tid; i < Zq * Zq; i += NTHREADS)
    sigp[i] = (i / Zq == i % Zq) ? 20.f : 0.f;
  for (int i = tid; i < Zq; i += NTHREADS) vmuprev[i] = 0.f;
  __syncthreads();

  // -------------------------- forward filter --------------------------
  for (int t = 0; t < Tq; ++t) {
    softmax16(alpha_b + t * Kq, vw);
    mix16(Acur, A_bank, vw, Zq * Zq);
    mix16(Cb, C_bank, vw, Aq * Zq);
    const int tn = (t + 1 < Tq) ? t + 1 : t;  // A_next; last repeats A_t
    softmax16(alpha_b + tn * Kq, vw);
    mix16(Anx, A_bank, vw, Zq * Zq);

    // T1 = C * sigp (32x64) -> tmp[0:2048)
    mm(Aq, Zq, Zq, Cb, Zq, false, sigp, Zq, false, tmp, Zq);
    // S = T1 * C^T + R (32x32) -> tmp[2048:3072)
    float* S = tmp + Aq * Zq;
    mm(Aq, Aq, Zq, tmp, Zq, false, Cb, Zq, true, S, Aq);
    if (tid < Aq) S[tid * Aq + tid] += 0.03f;
    __syncthreads();
    gj_inverse(S, Aq);                                // S <- S^-1
    // T2 = sigp * C^T (64x32) -> tmp[0:2048)
    mm(Zq, Aq, Zq, sigp, Zq, false, Cb, Zq, true, tmp, Aq);
    // Kg = T2 * S^-1 (64x32)
    mm(Zq, Aq, Aq, tmp, Aq, false, S, Aq, false, Kg, Aq);
    // Amu = A_t * mu_prev ; err = y - C*Amu
    matvec(Zq, Zq, Acur, Zq, vmuprev, vAmu);
    matvec(Aq, Zq, Cb, Zq, vAmu, vt);
    if (tid < Aq) verr[tid] = obs_b[t * Aq + tid] - vt[tid];
    __syncthreads();
    // mu_f = Amu + Kg*err
    matvec(Zq, Aq, Kg, Aq, verr, vt);
    if (tid < Zq) vmf[tid] = vAmu[tid] + vt[tid];
    __syncthreads();
    // IKC = I - Kg*C -> tmp ; sig_f = IKC * sigp
    mm(Zq, Zq, Aq, Kg, Aq, false, Cb, Zq, false, tmp, Zq);
    for (int i = tid; i < Zq * Zq; i += NTHREADS) {
      float v = -tmp[i];
      if (i / Zq == i % Zq) v += 1.f;
      tmp[i] = v;
    }
    __syncthreads();
    mm(Zq, Zq, Zq, tmp, Zq, false, sigp, Zq, false, sigf, Zq);
    // stage filtered results into d_out (overwritten by smoother pass)
    for (int i = tid; i < Zq; i += NTHREADS) mu_out[t * Zq + i] = vmf[i];
    for (int i = tid; i < Zq * Zq; i += NTHREADS)
      sig_out[(size_t)t * Zq * Zq + i] = sigf[i];
    __syncthreads();
    // predict: sigp = Anx*sigf*Anx^T + Q ; mu_prev = mu_f
    mm(Zq, Zq, Zq, Anx, Zq, false, sigf, Zq, false, tmp, Zq);
    mm(Zq, Zq, Zq, tmp, Zq, false, Anx, Zq, true, sigp, Zq);
    for (int i = tid; i < Zq * Zq; i += NTHREADS)
      if (i / Zq == i % Zq) sigp[i] += 0.08f;
    for (int i = tid; i < Zq; i += NTHREADS) vmuprev[i] = vmf[i];
    __syncthreads();
  }

  // -------------------------- backward smoother --------------------------
  for (int i = tid; i < Zq; i += NTHREADS) vmus[i] = vmf[i];
  for (int i = tid; i < Zq * Zq; i += NTHREADS) sigs[i] = sigf[i];
  __syncthreads();

  for (int t = Tq - 2; t >= 0; --t) {
    softmax16(alpha_b + (t + 1) * Kq, vw);
    mix16(Anx, A_bank, vw, Zq * Zq);
    // reload filtered t (still untouched at index t)
    for (int i = tid; i < Zq; i += NTHREADS) vmf[i] = mu_out[t * Zq + i];
    for (int i = tid; i < Zq * Zq; i += NTHREADS)
      sigf[i] = sig_out[(size_t)t * Zq * Zq + i];
    __syncthreads();
    // spn = Anx*sigf*Anx^T + Q -> sigp (recomputed: no HBM workspace needed)
    mm(Zq, Zq, Zq, Anx, Zq, false, sigf, Zq, false, tmp, Zq);
    mm(Zq, Zq, Zq, tmp, Zq, false, Anx, Zq, true, sigp, Zq);
    for (int i = tid; i < Zq * Zq; i += NTHREADS)
      if (i / Zq == i % Zq) sigp[i] += 0.08f;
    __syncthreads();
    // mpn = Anx * mu_f ; dmu = mu_s - mpn ; D = sig_s - spn (in place)
    matvec(Zq, Zq, Anx, Zq, vmf, vAmu);
    if (tid < Zq) verr[tid] = vmus[tid] - vAmu[tid];
    for (int i = tid; i < Zq * Zq; i += NTHREADS) sigs[i] -= sigp[i];
    __syncthreads();
    gj_inverse(sigp, Zq);  // sigp <- spn^-1
    // J = sigf * Anx^T * spn^-1  -> Acur (free during backward pass)
    mm(Zq, Zq, Zq, sigf, Zq, false, Anx, Zq, true, tmp, Zq);
    mm(Zq, Zq, Zq, tmp, Zq, false, sigp, Zq, false, Acur, Zq);
    // mu_s = mu_f + J*dmu
    matvec(Zq, Zq, Acur, Zq, verr, vt);
    if (tid < Zq) vmus[tid] = vmf[tid] + vt[tid];
    __syncthreads();
    // sig_s = sig_f + J*D*J^T
    mm(Zq, Zq, Zq, Acur, Zq, false, sigs, Zq, false, tmp, Zq);
    mm(Zq, Zq, Zq, tmp, Zq, false, Acur, Zq, true, sigs, Zq);
    for (int i = tid; i < Zq * Zq; i += NTHREADS) sigs[i] += sigf[i];
    __syncthreads();
    for (int i = tid; i < Zq; i += NTHREADS) mu_out[t * Zq + i] = vmus[i];
    for (int i = tid; i < Zq * Zq; i += NTHREADS)
      sig_out[(size_t)t * Zq * Zq + i] = sigs[i];
    __syncthreads();
  }
}

extern "C" void kernel_launch(void* const* d_in, const int* in_sizes, int n_in,
                              void* d_out, int out_size, void* d_ws, size_t ws_size,
                              hipStream_t stream) {
  (void)in_sizes; (void)n_in; (void)d_ws; (void)ws_size; (void)out_size;
  const float* obs    = (const float*)d_in[0];
  const float* alpha  = (const float*)d_in[1];
  const float* A_bank = (const float*)d_in[2];
  const float* C_bank = (const float*)d_in[3];
  float* out = (float*)d_out;
  const size_t shmem = (size_t)SMEM_FLOATS * sizeof(float);  // ~114 KB of 320 KB/WGP
  kvae_kernel<<<dim3(Bq), dim3(NTHREADS), shmem, stream>>>(obs, alpha, A_bank,
                                                           C_bank, out);
}